// GraphAttentionConv2d_17042430231095
// MI455X (gfx1250) — compile-verified
//
#include <hip/hip_runtime.h>
#include <math.h>

typedef __attribute__((ext_vector_type(2))) float v2f;
typedef __attribute__((ext_vector_type(8))) float v8f;

#define HEADS   2
#define OUTC    128
#define CIN     128
#define JTOT    256      // HEADS*OUTC
#define DHW     8192     // 8*32*32
#define BATCH   2
#define NNODES  16384    // BATCH*DHW

// ---------------------------------------------------------------- utilities
__device__ __forceinline__ void atomicMaxFloat(float* addr, float val) {
  // valid when slot initialized to -inf
  if (val >= 0.0f) atomicMax((int*)addr, __float_as_int(val));
  else             atomicMin((unsigned int*)addr, __float_as_uint(val));
}

// ---------------------------------------------------------------- init scratch
__global__ void k_init(float* __restrict__ acc, float* __restrict__ denom,
                       float* __restrict__ maxb, int accN, int n2) {
  int i = blockIdx.x * blockDim.x + threadIdx.x;
  if (i < accN) acc[i] = 0.0f;
  if (i < n2) { denom[i] = 0.0f; maxb[i] = -INFINITY; }
}

// ---------------------------------------------------------------- WMMA GEMM
// h[n][j] = sum_k xf[n][k] * lw[k][j] + lb[j]
// xf[n][k] = x[b][k][s], n = b*DHW + s.   One wave per 16x16 output tile.
__global__ void k_gemm(const float* __restrict__ x, const float* __restrict__ lw,
                       const float* __restrict__ lb, float* __restrict__ h) {
  int waveId = (blockIdx.x * blockDim.x + threadIdx.x) >> 5;
  int lane   = threadIdx.x & 31;
  int tileN  = waveId & 15;        // 256/16 column tiles
  int tileM  = waveId >> 4;        // 16384/16 row tiles
  int M0 = tileM * 16;
  int J0 = tileN * 16;

  // A fragment source: row m, columns k..k+3 split across lane halves
  int m = M0 + (lane & 15);
  int b = m >> 13;                 // DHW = 2^13
  int s = m & (DHW - 1);
  const float* xbase = x + (size_t)b * (CIN * DHW) + s;   // + k*DHW per channel
  int khalf = (lane >> 4) << 1;    // lanes 0-15 -> K+0/K+1 ; lanes 16-31 -> K+2/K+3

  // B fragment source: lw row k, column j (contiguous across lanes 0-15)
  int j = J0 + (lane & 15);
  const float* wbase = lw + j;

  v8f acc = {};
  #pragma unroll 4
  for (int k = 0; k < CIN; k += 4) {
    int ka = k + khalf;
    v2f a;  a.x  = xbase[(size_t)ka * DHW];       a.y  = xbase[(size_t)(ka + 1) * DHW];
    v2f bf; bf.x = wbase[(size_t)ka * JTOT];      bf.y = wbase[(size_t)(ka + 1) * JTOT];
    acc = __builtin_amdgcn_wmma_f32_16x16x4_f32(false, a, false, bf,
                                                (short)0, acc, false, false);
  }

  float bias = lb[j];
  int rowHalf = (lane >> 4) << 3;  // C/D: vgpr r -> row r (lanes 0-15) / r+8 (lanes 16-31)
  float* hb = h + (size_t)M0 * JTOT + j;
  #pragma unroll
  for (int r = 0; r < 8; ++r)
    hb[(size_t)(r + rowHalf) * JTOT] = acc[r] + bias;
}

// ---------------------------------------------------------------- edge pass 1
// one wave per (edge, head): logit = att . lrelu(h_src + h_dst, 0.2); segment max
__global__ void k_logits(const float* __restrict__ h, const int* __restrict__ ei,
                         const float* __restrict__ att, float* __restrict__ logits,
                         float* __restrict__ maxb, int E, int Etot) {
  int gw   = (blockIdx.x * blockDim.x + threadIdx.x) >> 5;
  int lane = threadIdx.x & 31;
  int e = gw >> 1, head = gw & 1;
  if (e >= Etot) return;
  int src, dst;
  if (e < E) {
    src = (int)((unsigned)ei[e]     % (unsigned)NNODES);
    dst = (int)((unsigned)ei[E + e] % (unsigned)NNODES);
  } else {
    src = dst = e - E;               // self loop
  }
  const float4* hs = (const float4*)(h + (size_t)src * JTOT + head * OUTC);
  const float4* hd = (const float4*)(h + (size_t)dst * JTOT + head * OUTC);
  const float4* at = (const float4*)(att + head * OUTC);
  float4 a = at[lane];
  float4 p = hs[lane];
  float4 q = hd[lane];
  float t, partial = 0.0f;
  t = p.x + q.x; t = (t > 0.0f) ? t : 0.2f * t; partial += a.x * t;
  t = p.y + q.y; t = (t > 0.0f) ? t : 0.2f * t; partial += a.y * t;
  t = p.z + q.z; t = (t > 0.0f) ? t : 0.2f * t; partial += a.z * t;
  t = p.w + q.w; t = (t > 0.0f) ? t : 0.2f * t; partial += a.w * t;
  #pragma unroll
  for (int off = 16; off > 0; off >>= 1)
    partial += __shfl_xor(partial, off, 32);
  if (lane == 0) {
    logits[gw] = partial;
    atomicMaxFloat(&maxb[dst * HEADS + head], partial);
  }
}

// ---------------------------------------------------------------- edge pass 2
__global__ void k_exp(const int* __restrict__ ei, float* __restrict__ logits,
                      const float* __restrict__ maxb, float* __restrict__ denom,
                      int E, int Etot) {
  int idx = blockIdx.x * blockDim.x + threadIdx.x;
  if (idx >= Etot * HEADS) return;
  int e = idx >> 1, head = idx & 1;
  int dst = (e < E) ? (int)((unsigned)ei[E + e] % (unsigned)NNODES) : e - E;
  float ex = __expf(logits[idx] - maxb[dst * HEADS + head]);
  logits[idx] = ex;
  atomicAdd(&denom[dst * HEADS + head], ex);
}

// ---------------------------------------------------------------- edge pass 3
__global__ void k_scatter(const float* __restrict__ h, const int* __restrict__ ei,
                          const float* __restrict__ logits, const float* __restrict__ denom,
                          float* __restrict__ acc, int E, int Etot) {
  int gw   = (blockIdx.x * blockDim.x + threadIdx.x) >> 5;
  int lane = threadIdx.x & 31;
  int e = gw >> 1, head = gw & 1;
  if (e >= Etot) return;
  int src, dst;
  if (e < E) {
    src = (int)((unsigned)ei[e]     % (unsigned)NNODES);
    dst = (int)((unsigned)ei[E + e] % (unsigned)NNODES);
  } else {
    src = dst = e - E;
  }
  float w = logits[gw] / denom[dst * HEADS + head];
  const float4* hs = (const float4*)(h + (size_t)src * JTOT + head * OUTC);
  float4 p = hs[lane];
  float* ab = acc + (size_t)dst * JTOT + head * OUTC + lane * 4;
  atomicAdd(ab + 0, p.x * w);
  atomicAdd(ab + 1, p.y * w);
  atomicAdd(ab + 2, p.z * w);
  atomicAdd(ab + 3, p.w * w);
}

// ---------------------------------------------------------------- head mean + bias
// writes directly in output layout [B][C][DHW]
__global__ void k_headmean(const float* __restrict__ acc, const float* __restrict__ gb,
                           float* __restrict__ out) {
  int idx = blockIdx.x * blockDim.x + threadIdx.x;
  if (idx >= NNODES * OUTC) return;
  int n = idx >> 7, c = idx & 127;
  float v = 0.5f * (acc[(size_t)n * JTOT + c] + acc[(size_t)n * JTOT + OUTC + c]) + gb[c];
  int b = n >> 13, s = n & (DHW - 1);
  out[(size_t)b * (OUTC * DHW) + (size_t)c * DHW + s] = v;
}

// ---------------------------------------------------------------- BN statistics
__global__ void k_bnstats(const float* __restrict__ out, float* __restrict__ stats) {
  __shared__ float s1[256];
  __shared__ float s2[256];
  int c = blockIdx.x;              // 0..127
  float sum = 0.0f, sq = 0.0f;
  for (int b = 0; b < BATCH; ++b) {
    const float* p = out + (size_t)b * (OUTC * DHW) + (size_t)c * DHW;
    for (int s = threadIdx.x; s < DHW; s += 256) {
      float v = p[s]; sum += v; sq += v * v;
    }
  }
  s1[threadIdx.x] = sum; s2[threadIdx.x] = sq;
  __syncthreads();
  for (int off = 128; off > 0; off >>= 1) {
    if ((int)threadIdx.x < off) {
      s1[threadIdx.x] += s1[threadIdx.x + off];
      s2[threadIdx.x] += s2[threadIdx.x + off];
    }
    __syncthreads();
  }
  if (threadIdx.x == 0) {
    float mean = s1[0] / (float)NNODES;
    float var  = s2[0] / (float)NNODES - mean * mean;
    stats[c]        = mean;
    stats[OUTC + c] = rsqrtf(var + 1e-5f);
  }
}

// ---------------------------------------------------------------- BN + LeakyReLU
__global__ void k_bnact(float* __restrict__ out, const float* __restrict__ stats,
                        const float* __restrict__ gamma, const float* __restrict__ beta) {
  int idx = blockIdx.x * blockDim.x + threadIdx.x;
  if (idx >= BATCH * OUTC * DHW) return;
  int c = (idx >> 13) & 127;
  float v = out[idx];
  v = (v - stats[c]) * stats[OUTC + c] * gamma[c] + beta[c];
  out[idx] = (v > 0.0f) ? v : 0.01f * v;
}

// ---------------------------------------------------------------- launcher
extern "C" void kernel_launch(void* const* d_in, const int* in_sizes, int n_in,
                              void* d_out, int out_size, void* d_ws, size_t ws_size,
                              hipStream_t stream) {
  (void)n_in; (void)out_size; (void)ws_size;
  const float* x     = (const float*)d_in[0];
  const int*   ei    = (const int*)d_in[1];
  const float* lw    = (const float*)d_in[2];
  const float* lb    = (const float*)d_in[3];
  const float* att   = (const float*)d_in[4];
  const float* gb    = (const float*)d_in[5];
  const float* gamma = (const float*)d_in[6];
  const float* beta  = (const float*)d_in[7];
  float* out = (float*)d_out;

  int E    = in_sizes[1] / 2;
  int Etot = E + NNODES;

  float* ws     = (float*)d_ws;
  float* h      = ws;                                   // N*256
  float* logits = h      + (size_t)NNODES * JTOT;       // Etot*2
  float* maxb   = logits + (size_t)Etot * HEADS;        // N*2
  float* denom  = maxb   + (size_t)NNODES * HEADS;      // N*2
  float* acc    = denom  + (size_t)NNODES * HEADS;      // N*256
  float* stats  = acc    + (size_t)NNODES * JTOT;       // 256

  int accN = NNODES * JTOT;
  k_init<<<(accN + 255) / 256, 256, 0, stream>>>(acc, denom, maxb, accN, NNODES * HEADS);

  int waves = (NNODES / 16) * (JTOT / 16);              // 16384 wave tiles
  k_gemm<<<waves / 8, 256, 0, stream>>>(x, lw, lb, h);

  int ewaves = Etot * HEADS;                            // one wave per (edge, head)
  k_logits<<<(ewaves + 7) / 8, 256, 0, stream>>>(h, ei, att, logits, maxb, E, Etot);

  k_exp<<<(Etot * HEADS + 255) / 256, 256, 0, stream>>>(ei, logits, maxb, denom, E, Etot);

  k_scatter<<<(ewaves + 7) / 8, 256, 0, stream>>>(h, ei, logits, denom, acc, E, Etot);

  k_headmean<<<(NNODES * OUTC + 255) / 256, 256, 0, stream>>>(acc, gb, out);

  k_bnstats<<<OUTC, 256, 0, stream>>>(out, stats);

  k_bnact<<<(BATCH * OUTC * DHW + 255) / 256, 256, 0, stream>>>(out, stats, gamma, beta);
}